// GatConvNN_2327872274906
// MI455X (gfx1250) — compile-verified
//
#include <hip/hip_runtime.h>
#include <hip/hip_bf16.h>

// ---------------------------------------------------------------------------
// GATConv on gfx1250 (MI455X), wave32.
//  h = x@W via V_WMMA_F32_16X16X4_F32 (fp32-exact matrix path)
//  segment softmax + scatter via L2 float atomics.
// ---------------------------------------------------------------------------

typedef __attribute__((ext_vector_type(2))) float v2f;
typedef __attribute__((ext_vector_type(8))) float v8f;

#define IN_C   256
#define OUT_C  64
#define HEADS  4
#define NEG_SLOPE 0.2f

// ----- order-preserving float<->uint mapping for atomic max on floats -------
__device__ __forceinline__ unsigned f2ord(float f) {
    unsigned u = __float_as_uint(f);
    return (u & 0x80000000u) ? ~u : (u | 0x80000000u);
}
__device__ __forceinline__ float ord2f(unsigned u) {
    return (u & 0x80000000u) ? __uint_as_float(u & 0x7FFFFFFFu)
                             : __uint_as_float(~u);
}

// ----- edge (src,dst) with implicit self-loops appended ---------------------
__device__ __forceinline__ void edge_sd(const int* __restrict__ ei, int E,
                                        int i, int& s, int& d) {
    if (i < E) { s = ei[i]; d = ei[E + i]; }
    else       { s = i - E; d = i - E; }
}

// ---------------------------------------------------------------------------
// 0) init: zero output accumulator, attention max (ordered -inf), denominator
// ---------------------------------------------------------------------------
__launch_bounds__(256)
__global__ void k_init(float* __restrict__ out, unsigned* __restrict__ mord,
                       float* __restrict__ denom, int Nn) {
    int tid = blockIdx.x * blockDim.x + threadIdx.x;
    if (tid < Nn * OUT_C) out[tid] = 0.0f;
    if (tid < Nn * HEADS) { mord[tid] = 0u; denom[tid] = 0.0f; }
}

// ---------------------------------------------------------------------------
// 1) GEMM: h[Nn,256] = x[Nn,256] @ W[256,256] with f32 WMMA.
//    One wave per 16 rows, 16 accumulator tiles (full 256 output cols).
// ---------------------------------------------------------------------------
__launch_bounds__(256, 1)
__global__ void k_gemm_wmma(const float* __restrict__ x,
                            const float* __restrict__ W,
                            float* __restrict__ h, int Nn) {
    const int lane = threadIdx.x & 31;
    const int wave = blockIdx.x * (blockDim.x >> 5) + (threadIdx.x >> 5);
    const int m0   = wave * 16;
    if (m0 >= Nn) return;                 // wave-uniform: EXEC all-1 below

    const int half = lane >> 4;           // 0: K pair {0,1}, 1: K pair {2,3}
    const int l16  = lane & 15;

    v8f acc[16] = {};                     // 16 N-tiles x 8 VGPRs = 128 VGPRs

    const float* __restrict__ arow  = x + (size_t)(m0 + l16) * IN_C + 2 * half;
    const float* __restrict__ wbase = W + (size_t)(2 * half) * (HEADS * OUT_C) + l16;

#pragma unroll 2
    for (int k = 0; k < IN_C; k += 4) {
        v2f a;
        a.x = arow[k];                    // (M=l16, K=k+2*half)
        a.y = arow[k + 1];                // (M=l16, K=k+2*half+1)
        __builtin_prefetch(&arow[k + 16], 0, 1);   // global_prefetch_b8

        const float* __restrict__ bk = wbase + (size_t)k * (HEADS * OUT_C);
#pragma unroll
        for (int t = 0; t < 16; ++t) {
            v2f b;
            b.x = bk[t * 16];                       // (K=k+2*half,   N=16t+l16)
            b.y = bk[t * 16 + HEADS * OUT_C];       // (K=k+2*half+1, N=16t+l16)
            acc[t] = __builtin_amdgcn_wmma_f32_16x16x4_f32(
                false, a, false, b, (short)0, acc[t], false, false);
        }
    }

    // C/D layout: VGPR v, lanes 0-15 -> M=m0+v, lanes 16-31 -> M=m0+8+v; N=16t+l16
#pragma unroll
    for (int t = 0; t < 16; ++t) {
#pragma unroll
        for (int v = 0; v < 8; ++v) {
            h[(size_t)(m0 + v + 8 * half) * (HEADS * OUT_C) + t * 16 + l16] = acc[t][v];
        }
    }
}

// ---------------------------------------------------------------------------
// 2) per-node attention dots: a_src[n,h], a_dst[n,h]
// ---------------------------------------------------------------------------
__launch_bounds__(256)
__global__ void k_attn_dots(const float* __restrict__ h,
                            const float* __restrict__ att_src,
                            const float* __restrict__ att_dst,
                            float* __restrict__ a_s, float* __restrict__ a_d,
                            int Nn) {
    int tid = blockIdx.x * blockDim.x + threadIdx.x;
    if (tid >= Nn * HEADS) return;
    int n = tid >> 2, hd = tid & 3;
    const float* __restrict__ hp = h + (size_t)n * (HEADS * OUT_C) + hd * OUT_C;
    const float* __restrict__ as = att_src + hd * OUT_C;
    const float* __restrict__ ad = att_dst + hd * OUT_C;
    float s = 0.0f, d = 0.0f;
#pragma unroll 8
    for (int c = 0; c < OUT_C; ++c) {
        float v = hp[c];
        s += v * as[c];
        d += v * ad[c];
    }
    a_s[tid] = s;
    a_d[tid] = d;
}

// ---------------------------------------------------------------------------
// 3) per-(edge,head) logits + segment max (atomic umax on ordered bits)
// ---------------------------------------------------------------------------
__launch_bounds__(256)
__global__ void k_edge_max(const int* __restrict__ ei, int E, int Nn,
                           const float* __restrict__ a_s,
                           const float* __restrict__ a_d,
                           float* __restrict__ e_out,
                           unsigned* __restrict__ mord) {
    int tid = blockIdx.x * blockDim.x + threadIdx.x;
    int tot = (E + Nn) * HEADS;
    if (tid >= tot) return;
    int i = tid >> 2, hd = tid & 3;
    int s, d;
    edge_sd(ei, E, i, s, d);
    float e = a_s[s * HEADS + hd] + a_d[d * HEADS + hd];
    e = (e > 0.0f) ? e : NEG_SLOPE * e;        // leaky_relu
    e_out[tid] = e;
    atomicMax(&mord[d * HEADS + hd], f2ord(e));
}

// ---------------------------------------------------------------------------
// 4) per-(edge,head) exp(e - m) + segment sum (denominator); ex stored in-place
// ---------------------------------------------------------------------------
__launch_bounds__(256)
__global__ void k_edge_exp(const int* __restrict__ ei, int E, int Nn,
                           float* __restrict__ e_io,
                           const unsigned* __restrict__ mord,
                           float* __restrict__ denom) {
    int tid = blockIdx.x * blockDim.x + threadIdx.x;
    int tot = (E + Nn) * HEADS;
    if (tid >= tot) return;
    int i = tid >> 2, hd = tid & 3;
    int s, d;
    edge_sd(ei, E, i, s, d);
    float m  = ord2f(mord[d * HEADS + hd]);
    float ex = __expf(e_io[tid] - m);
    e_io[tid] = ex;
    atomicAdd(&denom[d * HEADS + hd], ex);
}

// ---------------------------------------------------------------------------
// 5) weighted scatter: out[dst,c] += 0.25 * sum_h alpha_h * h[src,h,c]
//    one thread per (edge, channel); 4 heads folded -> single atomic per float
// ---------------------------------------------------------------------------
__launch_bounds__(256)
__global__ void k_edge_scatter(const int* __restrict__ ei, int E, int Nn,
                               const float* __restrict__ ex,
                               const float* __restrict__ denom,
                               const float* __restrict__ h,
                               float* __restrict__ out) {
    long long tid = (long long)blockIdx.x * blockDim.x + threadIdx.x;
    long long tot = (long long)(E + Nn) * OUT_C;
    if (tid >= tot) return;
    int i = (int)(tid >> 6);
    int c = (int)(tid & 63);
    int s, d;
    edge_sd(ei, E, i, s, d);
    const float* __restrict__ hs = h + (size_t)s * (HEADS * OUT_C) + c;
    float v = 0.0f;
#pragma unroll
    for (int hd = 0; hd < HEADS; ++hd) {
        float alpha = ex[(size_t)i * HEADS + hd] /
                      (denom[(size_t)d * HEADS + hd] + 1e-16f);
        v += alpha * hs[hd * OUT_C];
    }
    atomicAdd(&out[(size_t)d * OUT_C + c], 0.25f * v);   // mean over 4 heads
}

// ---------------------------------------------------------------------------
// 6) add bias
// ---------------------------------------------------------------------------
__launch_bounds__(256)
__global__ void k_bias(float* __restrict__ out, const float* __restrict__ bias,
                       int Nn) {
    int tid = blockIdx.x * blockDim.x + threadIdx.x;
    if (tid >= Nn * OUT_C) return;
    out[tid] += bias[tid & 63];
}

// ---------------------------------------------------------------------------
extern "C" void kernel_launch(void* const* d_in, const int* in_sizes, int n_in,
                              void* d_out, int out_size, void* d_ws, size_t ws_size,
                              hipStream_t stream) {
    const float* x       = (const float*)d_in[0];
    const int*   ei      = (const int*)  d_in[1];
    const float* W       = (const float*)d_in[2];
    const float* att_src = (const float*)d_in[3];
    const float* att_dst = (const float*)d_in[4];
    const float* bias    = (const float*)d_in[5];
    float*       out     = (float*)d_out;

    const int Nn = in_sizes[0] / IN_C;   // 50000
    const int E  = in_sizes[1] / 2;      // 800000
    const int NE = E + Nn;               // edges incl. self-loops

    // workspace carve (~68 MB)
    float*    h     = (float*)d_ws;                      // Nn*256
    float*    a_s   = h + (size_t)Nn * (HEADS * OUT_C);  // Nn*4
    float*    a_d   = a_s + (size_t)Nn * HEADS;          // Nn*4
    unsigned* mord  = (unsigned*)(a_d + (size_t)Nn * HEADS);  // Nn*4
    float*    denom = (float*)(mord + (size_t)Nn * HEADS);    // Nn*4
    float*    exbuf = denom + (size_t)Nn * HEADS;             // NE*4

    const int TB = 256;

    // 0) init accumulators
    k_init<<<(Nn * OUT_C + TB - 1) / TB, TB, 0, stream>>>(out, mord, denom, Nn);

    // 1) h = x @ W  (WMMA f32; one wave per 16 rows, 8 waves per block)
    {
        int waves  = (Nn + 15) / 16;
        int blocks = (waves + 7) / 8;
        k_gemm_wmma<<<blocks, TB, 0, stream>>>(x, W, h, Nn);
    }

    // 2) attention dot products
    k_attn_dots<<<(Nn * HEADS + TB - 1) / TB, TB, 0, stream>>>(
        h, att_src, att_dst, a_s, a_d, Nn);

    // 3) logits + segment max
    k_edge_max<<<((long long)NE * HEADS + TB - 1) / TB, TB, 0, stream>>>(
        ei, E, Nn, a_s, a_d, exbuf, mord);

    // 4) exp + denominator
    k_edge_exp<<<((long long)NE * HEADS + TB - 1) / TB, TB, 0, stream>>>(
        ei, E, Nn, exbuf, mord, denom);

    // 5) weighted scatter-accumulate
    k_edge_scatter<<<(unsigned)(((long long)NE * OUT_C + TB - 1) / TB), TB, 0, stream>>>(
        ei, E, Nn, exbuf, denom, h, out);

    // 6) bias
    k_bias<<<(Nn * OUT_C + TB - 1) / TB, TB, 0, stream>>>(out, bias, Nn);
}